// ARIGUserEncoder_88639535055482
// MI455X (gfx1250) — compile-verified
//
#include <hip/hip_runtime.h>
#include <math.h>
#include <stdint.h>

// Problem constants (match reference)
#define NB 2048
#define NT 200
#define ND 128
#define K_SHORT 5
#define LN_EPS 1e-5f
#define XSTR 130  // padded LDS row stride (floats): 8B-aligned rows for async B64,
                  // conflict-free banks for both row-wise and column-wise phases.

typedef __attribute__((ext_vector_type(2))) float v2f;
typedef __attribute__((ext_vector_type(8))) float v8f;
typedef __attribute__((ext_vector_type(2))) int   v2i;

#define AS1 __attribute__((address_space(1)))
#define AS3 __attribute__((address_space(3)))

#if defined(__AMDGCN__) && __has_builtin(__builtin_amdgcn_global_load_async_to_lds_b64)
#define USE_ASYNC_LDS 1
#else
#define USE_ASYNC_LDS 0
#endif

// ---------------- wave32 / block reductions (2 barriers instead of 8) -------
__device__ __forceinline__ float wred_add(float v) {
#pragma unroll
  for (int m = 16; m > 0; m >>= 1) v += __shfl_xor(v, m, 32);
  return v;
}
__device__ __forceinline__ float wred_max(float v) {
#pragma unroll
  for (int m = 16; m > 0; m >>= 1) v = fmaxf(v, __shfl_xor(v, m, 32));
  return v;
}
__device__ __forceinline__ float block_sum(float v, float* wr, float* res, int tid) {
  v = wred_add(v);
  if ((tid & 31) == 0) wr[tid >> 5] = v;
  __syncthreads();
  if (tid == 0) {
    float x = wr[0];
#pragma unroll
    for (int i = 1; i < 8; ++i) x += wr[i];
    *res = x;
  }
  __syncthreads();
  return *res;
}
__device__ __forceinline__ float block_max(float v, float* wr, float* res, int tid) {
  v = wred_max(v);
  if ((tid & 31) == 0) wr[tid >> 5] = v;
  __syncthreads();
  if (tid == 0) {
    float x = wr[0];
#pragma unroll
    for (int i = 1; i < 8; ++i) x = fmaxf(x, wr[i]);
    *res = x;
  }
  __syncthreads();
  return *res;
}

// ---------------- WMMA 128x128 matvec --------------------------------------
// vout = W * vin   (transposed == false), W row-major [128,128]
// vout = W^T * vin (transposed == true)
// One 16-wide output tile per wave; 32 chained V_WMMA_F32_16X16X4_F32.
// A (16x4 f32): lane0 = {K0,K1}, lane16 = {K2,K3}, others 0 (vector = row M=0).
// B (4x16): lane n (0..15) VGPR0/1 = K{2*hi, 2*hi+1}, hi = lane>>4.
// D row M=0 lands in VGPR0 of lanes 0..15.
__device__ __forceinline__ void matvec128_wmma(const float* __restrict__ W,
                                               const float* vin, float* vout,
                                               int wave, int lane, bool transposed) {
  const int nl = lane & 15;
  const int hi = lane >> 4;
  const int n  = (wave << 4) + nl;
  const bool l0  = (lane == 0);
  const bool l16 = (lane == 16);
  v8f acc = {0.f, 0.f, 0.f, 0.f, 0.f, 0.f, 0.f, 0.f};
#pragma unroll 4
  for (int k = 0; k < 32; ++k) {
    // Unconditional wave-uniform LDS broadcasts, then register selects
    // (v_cndmask) -> no EXEC-mask branching around the WMMA.
    const float vk0 = vin[4 * k + 0];
    const float vk1 = vin[4 * k + 1];
    const float vk2 = vin[4 * k + 2];
    const float vk3 = vin[4 * k + 3];
    float a0 = l0 ? vk0 : 0.f;  a0 = l16 ? vk2 : a0;
    float a1 = l0 ? vk1 : 0.f;  a1 = l16 ? vk3 : a1;
    v2f a = {a0, a1};
    float b0, b1;
    if (transposed) {
      b0 = W[(4 * k + 2 * hi) * ND + n];
      b1 = W[(4 * k + 2 * hi + 1) * ND + n];
    } else {
      b0 = W[n * ND + 4 * k + 2 * hi];
      b1 = W[n * ND + 4 * k + 2 * hi + 1];
    }
    v2f bb = {b0, b1};
    acc = __builtin_amdgcn_wmma_f32_16x16x4_f32(false, a, false, bb,
                                                (short)0, acc, false, false);
  }
  if (lane < 16) vout[n] = acc[0];
}

__launch_bounds__(256)
__global__ void arig_user_encoder_kernel(
    const float* __restrict__ hist,        // [B,T,D]
    const unsigned char* __restrict__ mask,// [B,T] (jax bool = 1 byte)
    const float* __restrict__ age,         // [B,T]
    const float* __restrict__ pop,         // [B,T]
    const float* __restrict__ dalpha,      // [1]
    const float* __restrict__ Wq,          // [D,D]
    const float* __restrict__ Wk,          // [D,D]
    const float* __restrict__ Wv,          // [D,D]
    const float* __restrict__ gw,          // [1,2]
    const float* __restrict__ gb,          // [1]
    const float* __restrict__ lng,         // [D]
    const float* __restrict__ lnb,         // [D]
    float* __restrict__ out)               // [B,D]
{
  __shared__ float xs[NT * XSTR];     // 104,000 B : x[b] tile (one HBM read)
  __shared__ float sc[NT];            // scores -> attention weights
  __shared__ float dlog[NT];          // log(decay + 1e-12)
  __shared__ float mf[NT];            // mask as float
  __shared__ float s_age[NT];
  __shared__ float s_pop[NT];
  __shared__ float vmean[ND], vq[ND], vqk[ND], vs[ND], vlt[ND];
  __shared__ float part[2][ND];
  __shared__ float s_wr[8];           // per-wave reduction slots
  __shared__ float s_res[1];          // reduction broadcast
  __shared__ float s_g[1];

  const int tid  = threadIdx.x;
  const int b    = blockIdx.x;
  const int lane = tid & 31;
  const int wave = tid >> 5;

  // Keep the three 64KB weight matrices hot in cache (global_prefetch path).
  {
    const int off = tid * 64;  // 256 threads * 256B = full 64KB per matrix
    __builtin_prefetch(Wq + off, 0, 3);
    __builtin_prefetch(Wk + off, 0, 3);
    __builtin_prefetch(Wv + off, 0, 3);
  }

  // ---- Phase 0: stage x[b] (200x128 fp32) into LDS; load per-t metadata ----
  const float* xb = hist + (size_t)b * NT * ND;
#if USE_ASYNC_LDS
  // ASYNCcnt-tracked DMA: no VGPR round-trip for the 102KB tile.
  for (int i = tid; i < NT * ND / 2; i += 256) {
    const int t  = i >> 6;          // i / (ND/2)
    const int d2 = (i & 63) << 1;   // (i % 64) * 2
    __builtin_amdgcn_global_load_async_to_lds_b64(
        (AS1 v2i*)(xb + t * ND + d2), (AS3 v2i*)(&xs[t * XSTR + d2]), 0, 0);
  }
#else
  for (int i = tid; i < NT * ND / 4; i += 256) {
    const int t  = i >> 5;          // i / (ND/4)
    const int d4 = (i & 31) << 2;   // (i % 32) * 4
    const float4 v4 = *(const float4*)(xb + t * ND + d4);
    float* dst = &xs[t * XSTR + d4];
    dst[0] = v4.x; dst[1] = v4.y; dst[2] = v4.z; dst[3] = v4.w;
  }
#endif
  const float alpha = log1pf(expf(dalpha[0])) + 1e-6f;
  if (tid < NT) {
    const float m = mask[(size_t)b * NT + tid] ? 1.0f : 0.0f;
    const float a = age[(size_t)b * NT + tid];
    mf[tid]    = m;
    s_age[tid] = a;
    s_pop[tid] = pop[(size_t)b * NT + tid];
    dlog[tid]  = logf(expf(-alpha * a) * m + 1e-12f);
  }
#if USE_ASYNC_LDS
#if __has_builtin(__builtin_amdgcn_s_wait_asynccnt)
  __builtin_amdgcn_s_wait_asynccnt(0);
#else
  asm volatile("s_wait_asynccnt 0x0" ::: "memory");
#endif
#endif
  __syncthreads();

  // ---- Phase 1: mask count + masked mean over T ----
  const float cntf = block_sum((tid < NT) ? mf[tid] : 0.f, s_wr, s_res, tid);

  {
    const int d = tid & 127, h = tid >> 7;  // 2 threads per dim, 100 t each
    float acc = 0.f;
#pragma unroll 4
    for (int t = h * 100; t < h * 100 + 100; ++t)
      acc += xs[t * XSTR + d] * mf[t];
    part[h][d] = acc;
  }
  __syncthreads();
  if (tid < ND) vmean[tid] = (part[0][tid] + part[1][tid]) / (cntf + 1e-6f);
  __syncthreads();

  // ---- Phase 2: q = Wq*mean ; qk = Wk^T*q   (WMMA matvecs) ----
  matvec128_wmma(Wq, vmean, vq, wave, lane, false);
  __syncthreads();
  matvec128_wmma(Wk, vq, vqk, wave, lane, true);
  __syncthreads();

  // ---- Phase 3: scores_t = qk.x_t / sqrt(D) + log(decay); softmax over t ----
  if (tid < NT) {
    const float* xr = &xs[tid * XSTR];
    float acc = 0.f;
#pragma unroll 8
    for (int d = 0; d < ND; ++d) acc += xr[d] * vqk[d];
    const float scv = acc * 0.08838834764831845f + dlog[tid];  // 1/sqrt(128)
    sc[tid] = (mf[tid] > 0.f) ? scv : -INFINITY;
  }
  __syncthreads();

  const float mx = block_max((tid < NT) ? sc[tid] : -INFINITY, s_wr, s_res, tid);
  float e = 0.f;
  if (tid < NT) e = (mf[tid] > 0.f) ? expf(sc[tid] - mx) : 0.f;
  const float esum = block_sum(e, s_wr, s_res, tid);
  if (tid < NT) sc[tid] = e / esum;  // normalized attention
  __syncthreads();

  // ---- Phase 4: s = sum_t attn_t * x_t ; long_term = Wv*s (WMMA) ----
  {
    const int d = tid & 127, h = tid >> 7;
    float acc = 0.f;
#pragma unroll 4
    for (int t = h * 100; t < h * 100 + 100; ++t)
      acc += sc[t] * xs[t * XSTR + d];
    part[h][d] = acc;
  }
  __syncthreads();
  if (tid < ND) vs[tid] = part[0][tid] + part[1][tid];
  __syncthreads();
  matvec128_wmma(Wv, vs, vlt, wave, lane, false);
  __syncthreads();

  // ---- Phase 5: last-K short-term pooling + sigmoid gate ----
  int cnt = (int)(cntf + 0.5f);
  if (cnt < 1) cnt = 1;
  int lo = cnt - K_SHORT;
  if (lo < 0) lo = 0;
  const float denom = (float)(cnt - lo);  // >= 1

  float stacc = 0.f;
  if (tid < ND) {
    for (int t = lo; t < cnt; ++t) stacc += xs[t * XSTR + tid];
  }
  if (tid == 128) {
    float pacc = 0.f, racc = 0.f;
    for (int t = lo; t < cnt; ++t) { pacc += s_pop[t]; racc += s_age[t]; }
    const float z = gw[0] * (pacc / denom) + gw[1] * (racc / denom) + gb[0];
    s_g[0] = 1.0f / (1.0f + expf(-z));
  }
  __syncthreads();

  const float g = s_g[0];
  const float userd =
      (tid < ND) ? (g * (stacc / denom) + (1.0f - g) * vlt[tid]) : 0.f;

  // ---- Phase 6: LayerNorm over D and store ----
  const float mu = block_sum(userd, s_wr, s_res, tid) * (1.0f / ND);
  const float dv = (tid < ND) ? (userd - mu) : 0.f;
  const float var = block_sum(dv * dv, s_wr, s_res, tid) * (1.0f / ND);

  if (tid < ND)
    out[(size_t)b * ND + tid] = dv * rsqrtf(var + LN_EPS) * lng[tid] + lnb[tid];
}

extern "C" void kernel_launch(void* const* d_in, const int* in_sizes, int n_in,
                              void* d_out, int out_size, void* d_ws, size_t ws_size,
                              hipStream_t stream) {
  const float*         hist   = (const float*)d_in[0];
  const unsigned char* mask   = (const unsigned char*)d_in[1];
  const float*         age    = (const float*)d_in[2];
  const float*         pop    = (const float*)d_in[3];
  const float*         dalpha = (const float*)d_in[4];
  const float*         Wq     = (const float*)d_in[5];
  const float*         Wk     = (const float*)d_in[6];
  const float*         Wv     = (const float*)d_in[7];
  const float*         gw     = (const float*)d_in[8];
  const float*         gb     = (const float*)d_in[9];
  const float*         lng    = (const float*)d_in[10];
  const float*         lnb    = (const float*)d_in[11];
  float*               out    = (float*)d_out;

  hipLaunchKernelGGL(arig_user_encoder_kernel, dim3(NB), dim3(256), 0, stream,
                     hist, mask, age, pop, dalpha, Wq, Wk, Wv, gw, gb, lng, lnb,
                     out);
}